// TinyModel_83021717832387
// MI455X (gfx1250) — compile-verified
//
#include <hip/hip_runtime.h>
#include <hip/hip_bf16.h>
#include <stdint.h>

// ---------------------------------------------------------------------------
// TinyModel (embed -> linear recurrence -> lm_head) for MI455X / gfx1250.
//
// Roofline: lm_head GEMM writes 8192x32000 fp32 = 1.05 GB -> HBM-bound
// (~50us @ 23.3 TB/s); its 537 GFLOP of fp32 WMMA is far below that, and the
// 131 MB weight matrix stays resident in the 192 MB L2. The sequential scan
// is restructured as a chunked affine scan (L=16, C=128): parallel local
// scans, a boundary scan using w_state^16 (computed by WMMA repeated
// squaring) with the power matrix resident in LDS (128.5KB/WG slice; CDNA5
// has 320KB/WGP), and a parallel correction pass.
//
// GEMM tiles are staged with GLOBAL_LOAD_ASYNC_TO_LDS_B128 (ASYNCcnt) into
// double-buffered LDS so WMMA issue overlaps the next tile's DMA.
// ---------------------------------------------------------------------------

#define VOCABN 32000
#define DIMN   1024
#define BATCHN 4
#define SEQN   2048
#define MTOK   (BATCHN * SEQN)   // 8192 tokens
#define LCH    16                // chunk length
#define NCH    (SEQN / LCH)      // 128 chunks

typedef float v2f __attribute__((ext_vector_type(2)));
typedef float v8f __attribute__((ext_vector_type(8)));
typedef int   v4i __attribute__((vector_size(4 * sizeof(int))));

#if __has_builtin(__builtin_amdgcn_global_load_async_to_lds_b128)
#define HAVE_ASYNC_LDS 1
#else
#define HAVE_ASYNC_LDS 0
#endif

// 16-byte global -> LDS copy: async DMA path on gfx1250, else via VGPRs.
// Builtin proto (from hipcc diagnostic): (AS1 v4i*, AS3 v4i*, imm, imm).
__device__ __forceinline__ void copy16(const float* g, float* l)
{
#if HAVE_ASYNC_LDS
    __builtin_amdgcn_global_load_async_to_lds_b128(
        (__attribute__((address_space(1))) v4i*)(uintptr_t)g,
        (__attribute__((address_space(3))) v4i*)(uint32_t)(uintptr_t)l,
        0, 0);
#else
    *(float4*)l = *(const float4*)g;
#endif
}

// Wait for this wave's async tile copies, then workgroup barrier.
__device__ __forceinline__ void wait_tiles()
{
#if HAVE_ASYNC_LDS
#if __has_builtin(__builtin_amdgcn_s_wait_asynccnt)
    __builtin_amdgcn_s_wait_asynccnt(0);
#else
    asm volatile("s_wait_asynccnt 0x0" ::: "memory");
#endif
#endif
    __syncthreads();
}

// ---------------------------------------------------------------------------
// WMMA fp32 GEMM:  C[m,n] = sum_k A[m,k] * (BT ? B[n,k] : B[k,n])
// Optional row gather on A (A row m = A[gather[m]]).
// Block tile 64(M) x 128(N) x 32(K); 8 waves in 2x4 grid; each wave owns a
// 32x32 region = 2x2 v_wmma_f32_16x16x4_f32 accumulators. Double-buffered
// LDS tiles filled by async global->LDS copies.
// ---------------------------------------------------------------------------
template <bool BT>
__global__ __launch_bounds__(256) void gemm_f32_wmma(
    const float* __restrict__ A, const float* __restrict__ B,
    float* __restrict__ C, const int* __restrict__ gather,
    int M, int N, int K)
{
    constexpr int BM = 64, BN = 128, BK = 32;
    constexpr int AS   = 36;   // padded LDS stride for A tile [m][k]
    constexpr int BS_T = 36;   // B tile [n][k] stride when BT
    constexpr int BS_N = 132;  // B tile [k][n] stride when !BT
    constexpr int BSZ  = BT ? (BN * BS_T) : (BK * BS_N);

    __shared__ float As[2][BM * AS];
    __shared__ float Bs[2][BSZ];

    const int tid  = threadIdx.x;
    const int w    = tid >> 5;
    const int lane = tid & 31;
    const int half = lane >> 4;   // lanes 16-31 hold the "+2K / +8M" halves
    const int lm   = lane & 15;
    const int wm   = w >> 2;      // 0..1  (M direction)
    const int wn   = w & 3;       // 0..3  (N direction)
    const long m0  = (long)blockIdx.x * BM;
    const long n0  = (long)blockIdx.y * BN;

    // ---- tile loaders (issue 16B copies; async path returns immediately) --
    auto loadA = [&](int pp, int kb) {
        const int r  = tid >> 2;
        const int fq = tid & 3;
        const long row = gather ? (long)gather[m0 + r] : (m0 + r);
        const float* src = A + row * K + kb;
        copy16(&src[fq * 8],     &As[pp][r * AS + fq * 8]);
        copy16(&src[fq * 8 + 4], &As[pp][r * AS + fq * 8 + 4]);
        if (kb + BK < K) __builtin_prefetch(&src[BK], 0, 3);  // global_prefetch_b8
    };
    auto loadB = [&](int pp, int kb) {
        if (BT) {   // B is [n][k] row-major (Linear weights): direct copy
            #pragma unroll
            for (int i = 0; i < 4; i++) {
                int fid = tid + 256 * i;
                int r = fid >> 3, fq = fid & 7;
                copy16(&B[(n0 + r) * (long)K + kb + fq * 4],
                       &Bs[pp][r * BS_T + fq * 4]);
            }
        } else {    // B is [k][n] row-major: keep layout, frag-read handles it
            #pragma unroll
            for (int i = 0; i < 4; i++) {
                int fid = tid + 256 * i;
                int r = fid >> 5, fq = fid & 31;
                copy16(&B[(long)(kb + r) * N + n0 + fq * 4],
                       &Bs[pp][r * BS_N + fq * 4]);
            }
        }
    };

    v8f acc[2][2] = {};

    loadA(0, 0);
    loadB(0, 0);
    wait_tiles();

    int pp = 0;
    for (int kb = 0; kb < K; kb += BK) {
        if (kb + BK < K) {          // overlap next tile's DMA with compute
            loadA(pp ^ 1, kb + BK);
            loadB(pp ^ 1, kb + BK);
        }
        // ---- compute: 8 K-steps of 4, 2x2 wmma tiles per wave ----
        #pragma unroll
        for (int kk = 0; kk < BK; kk += 4) {
            const int kA = kk + 2 * half;  // ISA 16x4 A layout: VGPR0=K0/K2, VGPR1=K1/K3
            v2f a[2], b[2];
            #pragma unroll
            for (int mi = 0; mi < 2; mi++) {
                int row = wm * 32 + mi * 16 + lm;
                a[mi].x = As[pp][row * AS + kA];
                a[mi].y = As[pp][row * AS + kA + 1];
            }
            #pragma unroll
            for (int ni = 0; ni < 2; ni++) {
                int col = wn * 32 + ni * 16 + lm;
                if (BT) {
                    b[ni].x = Bs[pp][col * BS_T + kA];
                    b[ni].y = Bs[pp][col * BS_T + kA + 1];
                } else {
                    b[ni].x = Bs[pp][kA * BS_N + col];
                    b[ni].y = Bs[pp][(kA + 1) * BS_N + col];
                }
            }
            #pragma unroll
            for (int mi = 0; mi < 2; mi++)
                #pragma unroll
                for (int ni = 0; ni < 2; ni++)
                    acc[mi][ni] = __builtin_amdgcn_wmma_f32_16x16x4_f32(
                        false, a[mi], false, b[ni], (short)0, acc[mi][ni],
                        false, false);
        }
        wait_tiles();
        pp ^= 1;
    }

    // ---- epilogue: C 16x16 layout -> VGPR r: M=r (lanes 0-15) / r+8 (16-31)
    #pragma unroll
    for (int mi = 0; mi < 2; mi++)
        #pragma unroll
        for (int ni = 0; ni < 2; ni++)
            #pragma unroll
            for (int r = 0; r < 8; r++) {
                long row = m0 + wm * 32 + mi * 16 + r + 8 * half;
                long col = n0 + wn * 32 + ni * 16 + lm;
                C[row * N + col] = acc[mi][ni][r];
            }
}

// ---------------------------------------------------------------------------
// Phase A: per-chunk local scan with zero init, hidden written in place
// over u. One WG per chunk; state [4][1024] ping-pongs in LDS; each of the
// 256 threads owns 4 columns x 4 batches (16 FMA chains) so the 4MB/step
// w_state stream from L2 is amortized 16x per load.
// ---------------------------------------------------------------------------
__global__ __launch_bounds__(256) void scan_local_kernel(
    const float* __restrict__ W,   // w_state [D][D], row e used as W^T column
    float* __restrict__ UH,        // [B][S][D]: u in, local hidden out
    float* __restrict__ Fc)        // [NCH][B][D]: chunk-final local states
{
    __shared__ float st[2][BATCHN][DIMN];
    const int c = blockIdx.x, tid = threadIdx.x;

    #pragma unroll
    for (int b = 0; b < BATCHN; b++)
        #pragma unroll
        for (int j = 0; j < 4; j++) st[0][b][tid + 256 * j] = 0.f;
    __syncthreads();

    const float* wrow[4];
    #pragma unroll
    for (int j = 0; j < 4; j++) wrow[j] = W + (long)(tid + 256 * j) * DIMN;

    int cur = 0;
    for (int k = 0; k < LCH; k++) {
        const int t = c * LCH + k;
        float acc[4][BATCHN];
        #pragma unroll
        for (int j = 0; j < 4; j++)
            #pragma unroll
            for (int b = 0; b < BATCHN; b++)
                acc[j][b] = UH[((long)b * SEQN + t) * DIMN + tid + 256 * j];

        for (int d = 0; d < DIMN; d += 4) {
            float4 s4[BATCHN];
            #pragma unroll
            for (int b = 0; b < BATCHN; b++)
                s4[b] = *(const float4*)&st[cur][b][d];   // LDS broadcast
            #pragma unroll
            for (int j = 0; j < 4; j++) {
                float4 w4 = *(const float4*)&wrow[j][d];
                #pragma unroll
                for (int b = 0; b < BATCHN; b++)
                    acc[j][b] += s4[b].x * w4.x + s4[b].y * w4.y +
                                 s4[b].z * w4.z + s4[b].w * w4.w;
            }
        }
        #pragma unroll
        for (int j = 0; j < 4; j++)
            #pragma unroll
            for (int b = 0; b < BATCHN; b++) {
                UH[((long)b * SEQN + t) * DIMN + tid + 256 * j] = acc[j][b];
                st[cur ^ 1][b][tid + 256 * j] = acc[j][b];
            }
        __syncthreads();
        cur ^= 1;
    }
    #pragma unroll
    for (int b = 0; b < BATCHN; b++)
        #pragma unroll
        for (int j = 0; j < 4; j++)
            Fc[((long)c * BATCHN + b) * DIMN + tid + 256 * j] =
                st[cur][b][tid + 256 * j];
}

// ---------------------------------------------------------------------------
// Agent-scope spin barrier for the boundary scan (32 co-resident WGs).
// ---------------------------------------------------------------------------
__device__ __forceinline__ void grid_barrier(unsigned* bar, unsigned nwg)
{
    __syncthreads();
    if (threadIdx.x == 0) {
        unsigned* cnt = bar;
        unsigned* gen = bar + 16;   // separate cacheline
        unsigned g = __hip_atomic_load(gen, __ATOMIC_ACQUIRE, __HIP_MEMORY_SCOPE_AGENT);
        unsigned old = __hip_atomic_fetch_add(cnt, 1u, __ATOMIC_ACQ_REL, __HIP_MEMORY_SCOPE_AGENT);
        if (old == nwg - 1u) {
            __hip_atomic_store(cnt, 0u, __ATOMIC_RELAXED, __HIP_MEMORY_SCOPE_AGENT);
            __hip_atomic_fetch_add(gen, 1u, __ATOMIC_ACQ_REL, __HIP_MEMORY_SCOPE_AGENT);
        } else {
            while (__hip_atomic_load(gen, __ATOMIC_ACQUIRE, __HIP_MEMORY_SCOPE_AGENT) == g)
                __builtin_amdgcn_s_sleep(1);
        }
    }
    __syncthreads();
}

// ---------------------------------------------------------------------------
// Phase B: boundary scan S_c = S_{c-1} @ (W^T)^L + F_{c-1}, i.e.
// S_c[b,e] = sum_d S_{c-1}[b,d] * Q[e,d] with Q = w_state^L.
// 32 WGs x 128 threads; WG g keeps Q rows [32g, 32g+32) resident in LDS
// (32 x 1028-padded = 128.5 KB) so each of the 127 serialized steps only
// exchanges the 16KB state through L2.
// ---------------------------------------------------------------------------
__global__ __launch_bounds__(128) void scan_boundary_kernel(
    const float* __restrict__ Q,    // w_state^LCH, [D][D]
    const float* __restrict__ Fc,   // [NCH][B][D]
    float* __restrict__ Sc,         // [NCH][B][D]; Sc[0] pre-zeroed
    unsigned* __restrict__ bar)     // pre-zeroed counters
{
    extern __shared__ float qs[];               // [32][1028]
    const int g = blockIdx.x, tid = threadIdx.x;
    const unsigned NW = gridDim.x;

    for (int i = 0; i < 64; i++) {              // one-time slice load
        int fid = tid + 128 * i;
        int r = fid >> 8, fq = fid & 255;
        *(float4*)&qs[r * 1028 + fq * 4] =
            *(const float4*)&Q[((long)g * 32 + r) * DIMN + fq * 4];
    }
    __syncthreads();

    const int el = tid >> 2, b = tid & 3;
    const int e = g * 32 + el;

    for (int c = 1; c < NCH; c++) {
        float acc = Fc[((long)(c - 1) * BATCHN + b) * DIMN + e];
        const float* sp = Sc + ((long)(c - 1) * BATCHN + b) * DIMN;
        for (int d = 0; d < DIMN; d += 4) {
            float4 s4 = *(const float4*)&sp[d];
            float4 q4 = *(const float4*)&qs[el * 1028 + d];
            acc += s4.x * q4.x + s4.y * q4.y + s4.z * q4.z + s4.w * q4.w;
        }
        Sc[((long)c * BATCHN + b) * DIMN + e] = acc;
        __threadfence();
        grid_barrier(bar, NW);
    }
}

// ---------------------------------------------------------------------------
// Phase C: hidden[cL+k] += S_c @ (W^T)^{k+1}; same structure as Phase A.
// ---------------------------------------------------------------------------
__global__ __launch_bounds__(256) void scan_correct_kernel(
    const float* __restrict__ W, const float* __restrict__ Sc,
    float* __restrict__ UH)
{
    const int c = blockIdx.x;
    if (c == 0) return;   // S_0 = 0 -> zero correction
    __shared__ float st[2][BATCHN][DIMN];
    const int tid = threadIdx.x;

    #pragma unroll
    for (int b = 0; b < BATCHN; b++)
        #pragma unroll
        for (int j = 0; j < 4; j++)
            st[0][b][tid + 256 * j] =
                Sc[((long)c * BATCHN + b) * DIMN + tid + 256 * j];
    __syncthreads();

    const float* wrow[4];
    #pragma unroll
    for (int j = 0; j < 4; j++) wrow[j] = W + (long)(tid + 256 * j) * DIMN;

    int cur = 0;
    for (int k = 0; k < LCH; k++) {
        const int t = c * LCH + k;
        float acc[4][BATCHN] = {};
        for (int d = 0; d < DIMN; d += 4) {
            float4 s4[BATCHN];
            #pragma unroll
            for (int b = 0; b < BATCHN; b++)
                s4[b] = *(const float4*)&st[cur][b][d];
            #pragma unroll
            for (int j = 0; j < 4; j++) {
                float4 w4 = *(const float4*)&wrow[j][d];
                #pragma unroll
                for (int b = 0; b < BATCHN; b++)
                    acc[j][b] += s4[b].x * w4.x + s4[b].y * w4.y +
                                 s4[b].z * w4.z + s4[b].w * w4.w;
            }
        }
        #pragma unroll
        for (int j = 0; j < 4; j++)
            #pragma unroll
            for (int b = 0; b < BATCHN; b++) {
                long idx = ((long)b * SEQN + t) * DIMN + tid + 256 * j;
                UH[idx] = UH[idx] + acc[j][b];
                st[cur ^ 1][b][tid + 256 * j] = acc[j][b];
            }
        __syncthreads();
        cur ^= 1;
    }
}

// ---------------------------------------------------------------------------
extern "C" void kernel_launch(void* const* d_in, const int* in_sizes, int n_in,
                              void* d_out, int out_size, void* d_ws, size_t ws_size,
                              hipStream_t stream)
{
    const int*   ids    = (const int*)  d_in[0];  // input_ids [B,S]
    const float* embedw = (const float*)d_in[1];  // [V,D]
    const float* w_in   = (const float*)d_in[2];  // [D,D]
    const float* w_st   = (const float*)d_in[3];  // [D,D]
    const float* lm_w   = (const float*)d_in[4];  // [V,D]
    float* out = (float*)d_out;                   // [B,S,V]

    float* u  = (float*)d_ws;                       // 8,388,608 f (u -> hidden, in place)
    float* qa = u  + (size_t)MTOK * DIMN;           // 1,048,576 f
    float* qb = qa + (size_t)DIMN * DIMN;           // 1,048,576 f
    float* Fc = qb + (size_t)DIMN * DIMN;           //   524,288 f
    float* Sc = Fc + (size_t)NCH * BATCHN * DIMN;   //   524,288 f
    unsigned* bar = (unsigned*)(Sc + (size_t)NCH * BATCHN * DIMN);

    (void)hipMemsetAsync(Sc,  0, (size_t)BATCHN * DIMN * sizeof(float), stream); // S_0 = 0
    (void)hipMemsetAsync(bar, 0, 128, stream);

    dim3 blk(256);
    // u = gather(embed, ids) @ w_in^T
    gemm_f32_wmma<true><<<dim3(MTOK / 64, DIMN / 128), blk, 0, stream>>>(
        embedw, w_in, u, ids, MTOK, DIMN, DIMN);
    // Q = w_state^16 by repeated squaring (plain A@B GEMMs)
    gemm_f32_wmma<false><<<dim3(DIMN / 64, DIMN / 128), blk, 0, stream>>>(
        w_st, w_st, qa, nullptr, DIMN, DIMN, DIMN);                 // W^2
    gemm_f32_wmma<false><<<dim3(DIMN / 64, DIMN / 128), blk, 0, stream>>>(
        qa, qa, qb, nullptr, DIMN, DIMN, DIMN);                     // W^4
    gemm_f32_wmma<false><<<dim3(DIMN / 64, DIMN / 128), blk, 0, stream>>>(
        qb, qb, qa, nullptr, DIMN, DIMN, DIMN);                     // W^8
    gemm_f32_wmma<false><<<dim3(DIMN / 64, DIMN / 128), blk, 0, stream>>>(
        qa, qa, qb, nullptr, DIMN, DIMN, DIMN);                     // W^16
    // chunked affine scan
    scan_local_kernel<<<NCH, blk, 0, stream>>>(w_st, u, Fc);
    scan_boundary_kernel<<<32, 128, 32 * 1028 * sizeof(float), stream>>>(
        qb, Fc, Sc, bar);
    scan_correct_kernel<<<NCH, blk, 0, stream>>>(w_st, Sc, u);
    // logits = hidden @ lm_head^T   (HBM-write-bound: ~1.05 GB)
    gemm_f32_wmma<true><<<dim3(MTOK / 64, VOCABN / 128), blk, 0, stream>>>(
        u, lm_w, out, nullptr, MTOK, VOCABN, DIMN);
    (void)in_sizes; (void)n_in; (void)out_size; (void)ws_size;
}